// ToRGBLayer_14474039788059
// MI455X (gfx1250) — compile-verified
//
#include <hip/hip_runtime.h>

typedef float v2f __attribute__((ext_vector_type(2)));
typedef float v8f __attribute__((ext_vector_type(8)));

namespace {
constexpr int   kB     = 8;
constexpr int   kCin   = 256;
constexpr int   kHW    = 256 * 256;   // 65536 pixels per (b, c) plane
constexpr int   kWdim  = 512;
constexpr int   kCout  = 3;
constexpr float kClamp = 256.0f;
constexpr float kInvSqrtWdim = 0.04419417382415922f;  // 1/sqrt(512)
constexpr float kInvSqrtCin  = 0.0625f;               // 1/sqrt(256)
constexpr int   kTilesPerWave = 4;    // 4 x 16-pixel tiles share one B operand
}

// ---------------------------------------------------------------------------
// Prep: styles[b,c] = (dot(w[b,:], affine_w[c,:]) / sqrt(512) + affine_b[c])
//                     / sqrt(256)
// Mpad[b][r][c] = (r < 3) ? weight[r][c] * styles[b,c] : 0
// Padded to 16 rows so WMMA B-operand loads are branch-free for all lanes.
// ---------------------------------------------------------------------------
__global__ __launch_bounds__(256) void torgb_prep_kernel(
    const float* __restrict__ w,         // [8][512]
    const float* __restrict__ weight,    // [3][256]
    const float* __restrict__ affine_w,  // [256][512]
    const float* __restrict__ affine_b,  // [256]
    float* __restrict__ Mpad)            // [8][16][256]
{
  const int b = blockIdx.x;
  const int c = threadIdx.x;

  const float* __restrict__ wrow = w + b * kWdim;
  const float* __restrict__ arow = affine_w + c * kWdim;

  float acc = 0.0f;
  #pragma unroll 8
  for (int k = 0; k < kWdim; ++k) acc = fmaf(wrow[k], arow[k], acc);

  const float style = (acc * kInvSqrtWdim + affine_b[c]) * kInvSqrtCin;

  float* __restrict__ Mb = Mpad + b * 16 * kCin;
  #pragma unroll
  for (int r = 0; r < 16; ++r) {
    Mb[r * kCin + c] = (r < kCout) ? weight[r * kCin + c] * style : 0.0f;
  }
}

// ---------------------------------------------------------------------------
// Main: per wave, FOUR consecutive 16-pixel tiles of one batch.
//   For each K-chunk of 4 channels:
//     one shared B operand (4x16 modulated weights, L2-resident)
//     four A operands (16 pixels x 4 channels of x, NT-streamed from HBM)
//     four v_wmma_f32_16x16x4_f32 into four accumulators.
//   A[m][k] = x[b, k, base+t*16+m]   (m = pixel = lane 0..15, K split by half)
//   B[k][n] = Mpad[b][n][k]          (n = out-channel = lane 0..15)
//   D[m][n] = out[b, n, base+t*16+m] -> lanes n<3 store 8 pixels each.
// 256 threads = 8 waves per block -> 512 pixels per block.
// ---------------------------------------------------------------------------
__global__ __launch_bounds__(256) void torgb_wmma_kernel(
    const float* __restrict__ x,     // [8][256][65536]
    const float* __restrict__ Mpad,  // [8][16][256]
    const float* __restrict__ bias,  // [3]
    float* __restrict__ out)         // [8][3][65536]
{
  const int lane = threadIdx.x & 31;
  const int wave = threadIdx.x >> 5;
  const int half = lane >> 4;    // K-half select (ISA lane-split layout)
  const int lm   = lane & 15;    // A: pixel row m;  B: out-channel col n

  // First of 4 consecutive tiles handled by this wave (all in same batch:
  // 4096 tiles/batch is divisible by 4).
  const int tile0 = (blockIdx.x * 8 + wave) * kTilesPerWave;
  const int b     = tile0 >> 12;             // tile / 4096
  const int base  = (tile0 & 4095) * 16;     // first pixel of tile 0

  const float* __restrict__ xb = x + (size_t)b * kCin * kHW + base + lm;
  const float* __restrict__ Mb = Mpad + (b * 16 + lm) * kCin;

  v8f acc[kTilesPerWave];
  #pragma unroll
  for (int t = 0; t < kTilesPerWave; ++t)
    acc[t] = (v8f){0.f, 0.f, 0.f, 0.f, 0.f, 0.f, 0.f, 0.f};

  #pragma unroll 2
  for (int k = 0; k < kCin; k += 4) {
    const int ka = k + half * 2;

    // Shared B operand: compiler merges the pair into one global_load_b64.
    v2f bm;
    bm.x = Mb[ka + 0];
    bm.y = Mb[ka + 1];

    #pragma unroll
    for (int t = 0; t < kTilesPerWave; ++t) {
      v2f a;
      // x is streamed exactly once (512 MB) -> non-temporal, keep L2 clean.
      a.x = __builtin_nontemporal_load(xb + (size_t)(ka + 0) * kHW + t * 16);
      a.y = __builtin_nontemporal_load(xb + (size_t)(ka + 1) * kHW + t * 16);
      // (neg_a, A, neg_b, B, c_mod, C, reuse_a, reuse_b)
      acc[t] = __builtin_amdgcn_wmma_f32_16x16x4_f32(
          false, a, false, bm, (short)0, acc[t], false, false);
    }
  }

  // D layout: VGPR j holds pixel (j + 8*half) for out-channel n = lm.
  if (lm < kCout) {
    const float bv = bias[lm];
    #pragma unroll
    for (int t = 0; t < kTilesPerWave; ++t) {
      float r[8];
      #pragma unroll
      for (int j = 0; j < 8; ++j)
        r[j] = fminf(fmaxf(acc[t][j] + bv, -kClamp), kClamp);

      float* __restrict__ op =
          out + ((size_t)b * kCout + lm) * kHW + base + t * 16 + half * 8;
      *(float4*)(op + 0) = make_float4(r[0], r[1], r[2], r[3]);
      *(float4*)(op + 4) = make_float4(r[4], r[5], r[6], r[7]);
    }
  }
}

// ---------------------------------------------------------------------------
extern "C" void kernel_launch(void* const* d_in, const int* in_sizes, int n_in,
                              void* d_out, int out_size, void* d_ws, size_t ws_size,
                              hipStream_t stream) {
  const float* x        = (const float*)d_in[0];  // (8, 256, 256, 256)
  const float* w        = (const float*)d_in[1];  // (8, 512)
  const float* weight   = (const float*)d_in[2];  // (3, 256, 1, 1)
  const float* bias     = (const float*)d_in[3];  // (3,)
  const float* affine_w = (const float*)d_in[4];  // (256, 512)
  const float* affine_b = (const float*)d_in[5];  // (256,)
  float* out = (float*)d_out;                     // (8, 3, 256, 256)

  float* Mpad = (float*)d_ws;  // 8*16*256 floats = 128 KB of scratch

  torgb_prep_kernel<<<kB, 256, 0, stream>>>(w, weight, affine_w, affine_b, Mpad);

  const int tiles  = kB * (kHW / 16);              // 32768 16-pixel tiles
  const int blocks = tiles / (8 * kTilesPerWave);  // 8 waves x 4 tiles each
  torgb_wmma_kernel<<<blocks, 256, 0, stream>>>(x, Mpad, bias, out);
}